// MatrixFactorization_11201274708684
// MI455X (gfx1250) — compile-verified
//
#include <hip/hip_runtime.h>
#include <hip/hip_bf16.h>

typedef float v2f __attribute__((ext_vector_type(2)));
typedef float v8f __attribute__((ext_vector_type(8)));

// One wave32 computes 16 outputs:
//   out[base + i] = dot(U[users[base+i]], V[items[base+i]]),  D = 64
// via 16 accumulating V_WMMA_F32_16X16X4_F32 ops; result is the diagonal of
// the 16x16 f32 accumulator. Memory-bound: each gathered 256B row is read
// exactly once (split 8B/16B-stride between half-wave partner lanes).
__global__ void __launch_bounds__(256)
mf_gather_dot_wmma(const int* __restrict__ users,
                   const int* __restrict__ items,
                   const float* __restrict__ U,
                   const float* __restrict__ V,
                   float* __restrict__ out,
                   int batch)
{
    const int lane  = threadIdx.x & 31;
    const int wave  = threadIdx.x >> 5;
    const int tile  = blockIdx.x * (blockDim.x >> 5) + wave;   // 16 outputs/tile
    const int base  = tile * 16;
    if (base >= batch) return;                                  // wave-uniform

    const int h   = lane & 15;                                  // row within tile
    const int sub = (lane >> 4) << 1;                           // 0 (K+0,1) or 2 (K+2,3)

    // Clamp for a possible ragged tail (batch=16384 -> never taken, but keeps
    // EXEC all-ones through the WMMAs instead of diverging).
    int bi = base + h;
    bi = bi < batch ? bi : batch - 1;

    const float* __restrict__ up = U + (long long)users[bi] * 64 + sub;
    const float* __restrict__ vp = V + (long long)items[bi] * 64 + sub;

    v8f acc = {};
#pragma unroll
    for (int k = 0; k < 16; ++k) {
        // A-chunk: U_rows[:, 4k : 4k+4]   (16x4, lane h has M=h)
        v2f a = *(const v2f*)(up + k * 4);
        // B-chunk: V_rows[:, 4k : 4k+4]^T (4x16, lane h has N=h)
        v2f b = *(const v2f*)(vp + k * 4);
        // args: (neg_a, A, neg_b, B, c_mod, C, reuse_a, reuse_b)
        acc = __builtin_amdgcn_wmma_f32_16x16x4_f32(
                  false, a, false, b, (short)0, acc, false, false);
    }

    // Diagonal extraction:
    //   i < 8 : lane i,     element i        (N=lane, M=vgpr)
    //   i >= 8: lane i+16,  element i-8      (N=lane-16, M=vgpr+8)
    const int e = lane & 7;
    float v = acc[0];
    v = (e == 1) ? acc[1] : v;
    v = (e == 2) ? acc[2] : v;
    v = (e == 3) ? acc[3] : v;
    v = (e == 4) ? acc[4] : v;
    v = (e == 5) ? acc[5] : v;
    v = (e == 6) ? acc[6] : v;
    v = (e == 7) ? acc[7] : v;

    const bool active = (lane < 8) || (lane >= 24);
    const int  row    = (lane < 8) ? lane : (lane - 16);
    if (active && (base + row) < batch)
        out[base + row] = v;
}

extern "C" void kernel_launch(void* const* d_in, const int* in_sizes, int n_in,
                              void* d_out, int out_size, void* d_ws, size_t ws_size,
                              hipStream_t stream)
{
    const int*   users = (const int*)d_in[0];
    const int*   items = (const int*)d_in[1];
    const float* U     = (const float*)d_in[2];
    const float* V     = (const float*)d_in[3];
    float*       out   = (float*)d_out;

    const int batch = in_sizes[0];          // 16384
    const int outsPerBlock = 16 * 8;        // 8 waves/block, 16 outputs/wave
    const int blocks = (batch + outsPerBlock - 1) / outsPerBlock;

    mf_gather_dot_wmma<<<blocks, 256, 0, stream>>>(users, items, U, V, out, batch);
}